// CA_MambaLayer_70944269795621
// MI455X (gfx1250) — compile-verified
//
#include <hip/hip_runtime.h>
#include <hip/hip_bf16.h>
#include <math.h>
#include <stdint.h>

// ---------------------------------------------------------------------------
// Mamba layer, MI455X (gfx1250, wave32).
// B=4, L=1024, DIM=256, D_INNER=512, D_STATE=64, DT_RANK=16, D_CONV=4.
// GEMM1/GEMM4: block-tiled WMMA GEMM, W-panel double-buffered in LDS via
//              global_load_async_to_lds_b128 (ASYNCcnt) -- 4 wmma per A-frag.
// GEMM2/GEMM3: simple one-tile-per-wave WMMA (odd shapes: N=80 / K=16).
// Scan: one wave32 per (batch, channel), 64 states in 2 VGPRs/lane.
// ---------------------------------------------------------------------------

#define BQ 4
#define LQ 1024
#define DIMQ 256
#define DINQ 512
#define DSTQ 64
#define DTRQ 16
#define ROWS (BQ * LQ)          // 4096
#define LDS_STRIDE 36           // floats: 16B-aligned rows, conflict-free banks

typedef __attribute__((ext_vector_type(2))) float v2f;
typedef __attribute__((ext_vector_type(8))) float v8f;

__device__ __forceinline__ float sigmoidf_(float v) {
    return 1.0f / (1.0f + __expf(-v));
}

// ---------------------------------------------------------------------------
// LayerNorm: one wave per row of 256, 8 floats/lane via two float4 loads.
// ---------------------------------------------------------------------------
__global__ __launch_bounds__(256) void ln_kernel(const float* __restrict__ x,
                                                 const float* __restrict__ gamma,
                                                 const float* __restrict__ beta,
                                                 float* __restrict__ xn) {
    const int lane = threadIdx.x & 31;
    const int wave = threadIdx.x >> 5;
    const int row  = blockIdx.x * 8 + wave;          // 0..4095
    const float4* xr = reinterpret_cast<const float4*>(x + (size_t)row * DIMQ);
    float4 a = xr[lane];
    float4 b = xr[lane + 32];
    float s  = a.x + a.y + a.z + a.w + b.x + b.y + b.z + b.w;
    float sq = a.x*a.x + a.y*a.y + a.z*a.z + a.w*a.w
             + b.x*b.x + b.y*b.y + b.z*b.z + b.w*b.w;
    #pragma unroll
    for (int off = 16; off; off >>= 1) {
        s  += __shfl_xor(s,  off, 32);
        sq += __shfl_xor(sq, off, 32);
    }
    const float mu  = s * (1.0f / DIMQ);
    const float var = sq * (1.0f / DIMQ) - mu * mu;
    const float rs  = rsqrtf(var + 1e-5f);
    const float4* g4 = reinterpret_cast<const float4*>(gamma);
    const float4* b4 = reinterpret_cast<const float4*>(beta);
    float4* o = reinterpret_cast<float4*>(xn + (size_t)row * DIMQ);
    float4 ga = g4[lane], gb = g4[lane + 32];
    float4 ba = b4[lane], bb = b4[lane + 32];
    float4 oa, ob;
    oa.x = (a.x - mu) * rs * ga.x + ba.x;  oa.y = (a.y - mu) * rs * ga.y + ba.y;
    oa.z = (a.z - mu) * rs * ga.z + ba.z;  oa.w = (a.w - mu) * rs * ga.w + ba.w;
    ob.x = (b.x - mu) * rs * gb.x + bb.x;  ob.y = (b.y - mu) * rs * gb.y + bb.y;
    ob.z = (b.z - mu) * rs * gb.z + bb.z;  ob.w = (b.w - mu) * rs * gb.w + bb.w;
    o[lane] = oa;
    o[lane + 32] = ob;
}

// ---------------------------------------------------------------------------
// Issue async DMA of a 64(n) x 32(k) W-panel chunk into LDS (all 256 threads,
// 2 x b128 per thread; 2 instructions -> +2 on the wave's ASYNCcnt).
// ---------------------------------------------------------------------------
__device__ __forceinline__ void issue_panel_async(const float* W, int ldw,
                                                  int n0, int kc,
                                                  uint32_t lds_base, int t) {
    #pragma unroll
    for (int s = 0; s < 2; ++s) {
        const int seg = t + s * 256;                 // 0..511
        const int row = seg >> 3;                    // 0..63
        const int sc  = seg & 7;                     // 0..7 (16B segment)
        const uint32_t voff  = (uint32_t)(((n0 + row) * ldw + kc + sc * 4) * 4);
        const uint32_t laddr = lds_base + (uint32_t)((row * LDS_STRIDE + sc * 4) * 4);
        asm volatile("global_load_async_to_lds_b128 %0, %1, %2"
                     :: "v"(laddr), "v"(voff), "s"(W) : "memory");
    }
}

// ---------------------------------------------------------------------------
// Tiled WMMA GEMM: C[m][n] = sum_k A[m][k] * W[n][k].
// Block = 128(m) x 64(n); 8 waves, each 16x64 (4 accumulator tiles, one
// shared A-fragment -> 4 wmma per k-step). W-panel double-buffered in LDS
// via async DMA. Requires M%128==0, N%64==0, K%32==0.
// ---------------------------------------------------------------------------
__global__ __launch_bounds__(256) void gemm_wmma_tiled(const float* __restrict__ A, int lda,
                                                       const float* __restrict__ W, int ldw,
                                                       float* __restrict__ C, int ldc,
                                                       int M, int N, int K) {
    __shared__ float ldsW[2][64 * LDS_STRIDE];       // 2 x 9 KB
    const int t    = threadIdx.x;
    const int lane = t & 31;
    const int w    = t >> 5;
    const int ncb  = N >> 6;
    const int brow = blockIdx.x / ncb;
    const int bcol = blockIdx.x - brow * ncb;
    const int n0   = bcol * 64;
    const int half = lane >> 4;
    const int l15  = lane & 15;

    const float* abase = A + (size_t)(brow * 128 + w * 16 + l15) * lda + half * 2;
    const uint32_t lb0 = (uint32_t)(uintptr_t)(&ldsW[0][0]);
    const uint32_t lb1 = (uint32_t)(uintptr_t)(&ldsW[1][0]);

    v8f acc0 = {}, acc1 = {}, acc2 = {}, acc3 = {};
    const int nc = K >> 5;

    issue_panel_async(W, ldw, n0, 0, lb0, t);        // prologue: chunk 0

    for (int c = 0; c < nc; ++c) {
        if (c + 1 < nc) {
            issue_panel_async(W, ldw, n0, (c + 1) * 32,
                              ((c + 1) & 1) ? lb1 : lb0, t);
            asm volatile("s_wait_asynccnt 0x2" ::: "memory");  // chunk c done
        } else {
            asm volatile("s_wait_asynccnt 0x0" ::: "memory");
        }
        __syncthreads();

        const float* ab = abase + c * 32;
        const float* lw = (c & 1) ? &ldsW[1][0] : &ldsW[0][0];
        #pragma unroll
        for (int k = 0; k < 32; k += 4) {
            v2f a  = *reinterpret_cast<const v2f*>(ab + k);
            v2f b0 = *reinterpret_cast<const v2f*>(lw + ( 0 + l15) * LDS_STRIDE + k + half * 2);
            v2f b1 = *reinterpret_cast<const v2f*>(lw + (16 + l15) * LDS_STRIDE + k + half * 2);
            v2f b2 = *reinterpret_cast<const v2f*>(lw + (32 + l15) * LDS_STRIDE + k + half * 2);
            v2f b3 = *reinterpret_cast<const v2f*>(lw + (48 + l15) * LDS_STRIDE + k + half * 2);
            acc0 = __builtin_amdgcn_wmma_f32_16x16x4_f32(false, a, false, b0, (short)0, acc0, false, false);
            acc1 = __builtin_amdgcn_wmma_f32_16x16x4_f32(false, a, false, b1, (short)0, acc1, false, false);
            acc2 = __builtin_amdgcn_wmma_f32_16x16x4_f32(false, a, false, b2, (short)0, acc2, false, false);
            acc3 = __builtin_amdgcn_wmma_f32_16x16x4_f32(false, a, false, b3, (short)0, acc3, false, false);
        }
        __syncthreads();                              // protect buffer reuse
    }

    const int m0 = brow * 128 + w * 16 + half * 8;
    #pragma unroll
    for (int j = 0; j < 8; ++j) {
        float* crow = C + (size_t)(m0 + j) * ldc + n0 + l15;
        crow[ 0] = acc0[j];
        crow[16] = acc1[j];
        crow[32] = acc2[j];
        crow[48] = acc3[j];
    }
}

// ---------------------------------------------------------------------------
// Simple WMMA GEMM (one 16x16 tile per wave) for odd shapes (N=80 / K=16).
// EPI: 0 = none, 1 = bias + softplus.
// ---------------------------------------------------------------------------
template <int EPI>
__global__ __launch_bounds__(256) void gemm_wmma(const float* __restrict__ A, int lda,
                                                 const float* __restrict__ W, int ldw,
                                                 float* __restrict__ C, int ldc,
                                                 const float* __restrict__ bias,
                                                 int M, int N, int K) {
    const int lane = threadIdx.x & 31;
    const int wave = threadIdx.x >> 5;
    const int tile = blockIdx.x * 8 + wave;
    const int ntn  = N >> 4;
    const int tm   = tile / ntn;
    const int tn   = tile - tm * ntn;
    if (tm * 16 >= M) return;                        // wave-uniform guard

    const int half = lane >> 4;
    const int l15  = lane & 15;
    const float* arow = A + (size_t)(tm * 16 + l15) * lda + half * 2;
    const float* wrow = W + (size_t)(tn * 16 + l15) * ldw + half * 2;

    v8f acc = {};
    for (int k = 0; k < K; k += 4) {
        v2f a = *reinterpret_cast<const v2f*>(arow + k);
        v2f b = *reinterpret_cast<const v2f*>(wrow + k);
        acc = __builtin_amdgcn_wmma_f32_16x16x4_f32(
            false, a, false, b, (short)0, acc, false, false);
    }

    const int m0 = tm * 16 + half * 8;
    const int n  = tn * 16 + l15;
    float bv = (EPI == 1) ? bias[n] : 0.0f;
    #pragma unroll
    for (int j = 0; j < 8; ++j) {
        float v = acc[j];
        if (EPI == 1) {                              // bias + softplus
            v += bv;
            v = (v > 20.0f) ? v : log1pf(__expf(v));
        }
        C[(size_t)(m0 + j) * ldc + n] = v;
    }
}

// ---------------------------------------------------------------------------
// Depthwise causal conv (k=4) + bias + SiLU.  u lives in xz cols [0,512).
// ---------------------------------------------------------------------------
__global__ __launch_bounds__(256) void conv_silu_kernel(const float* __restrict__ xz,
                                                        const float* __restrict__ cw,
                                                        const float* __restrict__ cb,
                                                        float* __restrict__ up) {
    const int idx = blockIdx.x * 256 + threadIdx.x;  // over ROWS*DINQ
    const int c = idx & (DINQ - 1);
    const int r = idx >> 9;                          // b*L + l
    const int l = r & (LQ - 1);
    const float4 w = reinterpret_cast<const float4*>(cw)[c]; // w[c][0..3]
    const float* u = xz + (size_t)r * (2 * DINQ) + c;
    float acc = cb[c] + w.w * u[0];                  // tap j=3 -> in[l]
    if (l >= 1) acc += w.z * u[-(2 * DINQ)];
    if (l >= 2) acc += w.y * u[-(4 * DINQ)];
    if (l >= 3) acc += w.x * u[-(6 * DINQ)];
    up[idx] = acc * sigmoidf_(acc);                  // SiLU
}

// ---------------------------------------------------------------------------
// Selective scan. One wave32 per (batch, channel): 64 states = 2 f32/lane.
// ---------------------------------------------------------------------------
__global__ __launch_bounds__(256) void scan_kernel(const float* __restrict__ dt,
                                                   const float* __restrict__ up,
                                                   const float* __restrict__ dbl,
                                                   const float* __restrict__ Csa,
                                                   const float* __restrict__ A_log,
                                                   float* __restrict__ ys) {
    const int lane = threadIdx.x & 31;
    const int wave = threadIdx.x >> 5;
    const int gw = blockIdx.x * 8 + wave;            // 0..2047
    const int b  = gw >> 9;                          // batch
    const int d  = gw & (DINQ - 1);                  // channel
    const int n0 = lane * 2;

    const float A0 = -__expf(A_log[(size_t)d * DSTQ + n0]);
    const float A1 = -__expf(A_log[(size_t)d * DSTQ + n0 + 1]);
    float h0 = 0.0f, h1 = 0.0f;

    for (int t = 0; t < LQ; ++t) {
        const size_t r = (size_t)b * LQ + t;
        const float dtv = dt[r * DINQ + d];
        const float uv  = up[r * DINQ + d];
        const v2f Bn = *reinterpret_cast<const v2f*>(dbl + r * 80 + DTRQ + n0);
        const v2f Cn = *reinterpret_cast<const v2f*>(Csa + r * DSTQ + n0);
        if (t + 1 < LQ) {                            // gfx1250 prefetch path
            __builtin_prefetch(dbl + (r + 1) * 80 + DTRQ + n0, 0, 1);
            __builtin_prefetch(Csa + (r + 1) * DSTQ + n0, 0, 1);
        }
        const float du = dtv * uv;
        h0 = __expf(dtv * A0) * h0 + du * Bn.x;
        h1 = __expf(dtv * A1) * h1 + du * Bn.y;
        float p = h0 * Cn.x + h1 * Cn.y;
        #pragma unroll
        for (int off = 16; off; off >>= 1) p += __shfl_xor(p, off, 32);
        if (lane == 0) ys[r * DINQ + d] = p;
    }
}

// ---------------------------------------------------------------------------
// Gate: yf = (ys + u*D) * SiLU(z);  z lives in xz cols [512,1024).
// ---------------------------------------------------------------------------
__global__ __launch_bounds__(256) void gate_kernel(const float* __restrict__ ys,
                                                   const float* __restrict__ up,
                                                   const float* __restrict__ Dp,
                                                   const float* __restrict__ xz,
                                                   float* __restrict__ yf) {
    const int idx = blockIdx.x * 256 + threadIdx.x;  // over ROWS*DINQ
    const int c = idx & (DINQ - 1);
    const size_t r = (size_t)(idx >> 9);
    const float z = xz[r * (2 * DINQ) + DINQ + c];
    yf[idx] = (ys[idx] + up[idx] * Dp[c]) * (z * sigmoidf_(z));
}

// ---------------------------------------------------------------------------
extern "C" void kernel_launch(void* const* d_in, const int* in_sizes, int n_in,
                              void* d_out, int out_size, void* d_ws, size_t ws_size,
                              hipStream_t stream) {
    const float* x      = (const float*)d_in[0];
    const float* C_SA   = (const float*)d_in[1];
    const float* gamma  = (const float*)d_in[2];
    const float* beta   = (const float*)d_in[3];
    const float* W_in   = (const float*)d_in[4];   // (1024, 256)
    const float* conv_w = (const float*)d_in[5];   // (512, 1, 4)
    const float* conv_b = (const float*)d_in[6];
    const float* W_x    = (const float*)d_in[7];   // (80, 512)
    const float* W_dt   = (const float*)d_in[8];   // (512, 16)
    const float* b_dt   = (const float*)d_in[9];
    const float* A_log  = (const float*)d_in[10];  // (512, 64)
    const float* Dp     = (const float*)d_in[11];
    const float* W_out  = (const float*)d_in[12];  // (256, 512)
    float* out = (float*)d_out;

    // Workspace layout (floats), total ~55.8 MB.
    float* ws  = (float*)d_ws;
    float* xn  = ws;                            // 4096*256
    float* xz  = xn  + (size_t)ROWS * DIMQ;     // 4096*1024
    float* up  = xz  + (size_t)ROWS * 2 * DINQ; // 4096*512
    float* dbl = up  + (size_t)ROWS * DINQ;     // 4096*80
    float* dtb = dbl + (size_t)ROWS * 80;       // 4096*512
    float* ys  = dtb + (size_t)ROWS * DINQ;     // 4096*512
    float* yf  = ys  + (size_t)ROWS * DINQ;     // 4096*512

    // 1. LayerNorm
    ln_kernel<<<ROWS / 8, 256, 0, stream>>>(x, gamma, beta, xn);

    // 2. xz = xn @ W_in.T    (M=4096, N=1024, K=256): tiled async WMMA
    gemm_wmma_tiled<<<(ROWS / 128) * (1024 / 64), 256, 0, stream>>>(
        xn, DIMQ, W_in, DIMQ, xz, 2 * DINQ, ROWS, 2 * DINQ, DIMQ);

    // 3. depthwise conv + SiLU
    conv_silu_kernel<<<(ROWS * DINQ) / 256, 256, 0, stream>>>(xz, conv_w, conv_b, up);

    // 4. dbl = u @ W_x.T     (M=4096, N=80, K=512): simple WMMA
    gemm_wmma<0><<<(ROWS / 16) * (80 / 16) / 8, 256, 0, stream>>>(
        up, DINQ, W_x, DINQ, dbl, 80, nullptr, ROWS, 80, DINQ);

    // 5. dt = softplus(dbl[:, :16] @ W_dt.T + b_dt)  (M=4096, N=512, K=16)
    gemm_wmma<1><<<(ROWS / 16) * (DINQ / 16) / 8, 256, 0, stream>>>(
        dbl, 80, W_dt, DTRQ, dtb, DINQ, b_dt, ROWS, DINQ, DTRQ);

    // 6. selective scan (2048 waves, state in registers)
    scan_kernel<<<(BQ * DINQ) / 8, 256, 0, stream>>>(dtb, up, dbl, C_SA, A_log, ys);

    // 7. gate
    gate_kernel<<<(ROWS * DINQ) / 256, 256, 0, stream>>>(ys, up, Dp, xz, yf);

    // 8. out = yf @ W_out.T  (M=4096, N=256, K=512): tiled async WMMA
    gemm_wmma_tiled<<<(ROWS / 128) * (DIMQ / 64), 256, 0, stream>>>(
        yf, DINQ, W_out, DINQ, out, DIMQ, ROWS, DIMQ, DINQ);
}